// CaTentLayer_66374424593116
// MI455X (gfx1250) — compile-verified
//
#include <hip/hip_runtime.h>

#define T_BINS   16384
#define NCH      1024
#define FW       128
#define CL       64                 // channels per block tile
#define TT       64                 // time outputs per tile
#define TG       8                  // sequential tiles per block (ring-pipelined)
#define RING     256                // ring rows (pow2; >= 191 live + 64 prefetch)
#define RR0      (TT + FW - 1)      // 191 rows for the initial fill
#define KB       8                  // k blocking
#define NTHREADS 256

#if defined(__has_builtin)
#if __has_builtin(__builtin_amdgcn_global_load_async_to_lds_b128)
#define HAVE_ASYNC_LDS 1
#endif
#if __has_builtin(__builtin_amdgcn_s_wait_asynccnt)
#define HAVE_WAIT_ASYNC 1
#endif
#endif

typedef int v4i __attribute__((vector_size(16)));
typedef __attribute__((address_space(1))) v4i* gv4_t;   // global v4i32*
typedef __attribute__((address_space(3))) v4i* lv4_t;   // LDS v4i32*
typedef float v2f __attribute__((ext_vector_type(2)));

__device__ __forceinline__ void stage16(const float* g, float* l) {
#ifdef HAVE_ASYNC_LDS
  // async DMA: global -> LDS, 16 bytes per lane, tracked by ASYNCcnt
  float* gm = const_cast<float*>(g);
  __builtin_amdgcn_global_load_async_to_lds_b128((gv4_t)gm, (lv4_t)l, 0, 0);
#else
  const float4 v = *(const float4*)g;
  *(float4*)l = v;
#endif
}

__device__ __forceinline__ void wait_async0() {
#ifdef HAVE_WAIT_ASYNC
  __builtin_amdgcn_s_wait_asynccnt(0);
#elif defined(HAVE_ASYNC_LDS)
  asm volatile("s_wait_asynccnt 0" ::: "memory");
#endif
}

// packed fp32 FMA on a channel pair: d = a*b + c (component-wise)
__device__ __forceinline__ v2f pk_fma(v2f a, v2f b, v2f c) {
  v2f d;
  asm("v_pk_fma_f32 %0, %1, %2, %3" : "=v"(d) : "v"(a), "v"(b), "v"(c));
  return d;
}

__global__ __launch_bounds__(NTHREADS) void catent_dwconv_kernel(
    const float* __restrict__ x,   // [T, NC]
    const float* __restrict__ W,   // [FW, NC]
    const float* __restrict__ b,   // [NC]
    float* __restrict__ out) {     // [T, NC]
  __shared__ alignas(16) float xs[RING][CL];  // ring of staged x rows (64 KB)
  __shared__ alignas(16) float fl[FW][CL];    // normalized filters (32 KB)
  __shared__ float rn[CL];

  const int tid = threadIdx.x;
  const int ct  = blockIdx.x & (NCH / CL - 1);   // 16 channel tiles
  const int bt  = blockIdx.x >> 4;               // 32 time groups
  const int c0  = ct * CL;
  const int s0  = bt * (TG * TT);

  // ---- stage raw W tile via async DMA: 128x64 floats = 2048 16B chunks ----
  for (int idx = tid; idx < (FW * CL) / 4; idx += NTHREADS) {
    const int r  = idx >> 4;
    const int ch = (idx & 15) << 2;
    stage16(&W[r * NCH + c0 + ch], &fl[r][ch]);
  }
  // ---- initial ring fill: rows g = s0-63 .. s0+127 (191 rows) ----
  for (int idx = tid; idx < (RR0 * CL) / 4; idx += NTHREADS) {
    const int r    = idx >> 4;
    const int ch   = (idx & 15) << 2;
    const int g    = s0 - (FW / 2 - 1) + r;      // s0 - 63 + r
    const int slot = g & (RING - 1);             // correct for negative g too
    if (g >= 0 && g < T_BINS) {
      stage16(&x[g * NCH + c0 + ch], &xs[slot][ch]);
    } else {
      *(float4*)&xs[slot][ch] = make_float4(0.f, 0.f, 0.f, 0.f);
    }
  }
  wait_async0();
  __syncthreads();

  // ---- per-channel 1/L2norm of the filter column ----
  if (tid < CL) {
    float s = 0.f;
#pragma unroll 8
    for (int k = 0; k < FW; ++k) {
      const float w = fl[k][tid];
      s = fmaf(w, w, s);
    }
    rn[tid] = 1.f / fmaxf(sqrtf(s), 1e-8f);
  }
  __syncthreads();
  // ---- normalize + ReLU (pos_constraint) in place ----
  for (int idx = tid; idx < FW * CL; idx += NTHREADS) {
    const int r  = idx >> 6;
    const int ch = idx & (CL - 1);
    fl[r][ch] = fmaxf(0.f, fl[r][ch] * rn[ch]);
  }
  __syncthreads();

  const int p  = tid & 31;            // channel pair (2 channels)
  const int ty = tid >> 5;            // 0..7 time groups
  const int j0 = ty * 8;
  const v2f bb = *(const v2f*)&b[c0 + 2 * p];

  for (int it = 0; it < TG; ++it) {
    const int S = s0 + it * TT;

    // ---- async prefetch of the next tile's 64 new rows (disjoint slots) ----
    if (it + 1 < TG) {
#pragma unroll
      for (int q = 0; q < (TT * CL) / 4 / NTHREADS; ++q) {   // 4 chunks/thread
        const int idx  = tid + q * NTHREADS;
        const int r    = idx >> 4;
        const int ch   = (idx & 15) << 2;
        const int g    = S + TT + FW / 2 + r;                // S + 128 + r
        const int slot = g & (RING - 1);
        if (g < T_BINS) {
          stage16(&x[g * NCH + c0 + ch], &xs[slot][ch]);
        } else {
          *(float4*)&xs[slot][ch] = make_float4(0.f, 0.f, 0.f, 0.f);
        }
      }
    }

    // ---- compute this tile while the DMA runs ----
    v2f acc[8];
#pragma unroll
    for (int i = 0; i < 8; ++i) acc[i] = (v2f)(0.f);

    const int roff = S - (FW / 2 - 1) + j0;        // ring row offset
    for (int k0 = 0; k0 < FW; k0 += KB) {
      v2f fk[KB];
#pragma unroll
      for (int u = 0; u < KB; ++u)
        fk[u] = *(const v2f*)&fl[k0 + u][2 * p];

      v2f xw[8 + KB - 1];
#pragma unroll
      for (int w = 0; w < 8 + KB - 1; ++w)
        xw[w] = *(const v2f*)&xs[(roff + k0 + w) & (RING - 1)][2 * p];

#pragma unroll
      for (int u = 0; u < KB; ++u) {
#pragma unroll
        for (int i = 0; i < 8; ++i)
          acc[i] = pk_fma(xw[i + u], fk[u], acc[i]);
      }
    }

    // ---- bias + circular roll(+64) fused into the store ----
#pragma unroll
    for (int i = 0; i < 8; ++i) {
      const int s = S + j0 + i;
      const int d = (s + FW / 2) & (T_BINS - 1);
      const v2f v = acc[i] + bb;
      *(v2f*)&out[d * NCH + c0 + 2 * p] = v;
    }

    wait_async0();
    __syncthreads();
  }
}

extern "C" void kernel_launch(void* const* d_in, const int* in_sizes, int n_in,
                              void* d_out, int out_size, void* d_ws, size_t ws_size,
                              hipStream_t stream) {
  (void)in_sizes; (void)n_in; (void)out_size; (void)d_ws; (void)ws_size;
  const float* x = (const float*)d_in[0];
  const float* W = (const float*)d_in[1];
  const float* b = (const float*)d_in[2];
  float* out = (float*)d_out;

  const int nblocks = (T_BINS / (TG * TT)) * (NCH / CL);   // 32 * 16 = 512
  catent_dwconv_kernel<<<nblocks, NTHREADS, 0, stream>>>(x, W, b, out);
}